// CF_50362786513123
// MI455X (gfx1250) — compile-verified
//
#include <hip/hip_runtime.h>
#include <math.h>

// Problem constants (from reference): D=20, 2B=131072 rows, B=65536 pairs.
#define DD 20
#define BLOCK 256                            // one row per thread
#define EPS_E_FLOATS (BLOCK * DD)            // 5120 floats = 20 KB LDS per block
#define EPS_E_B128_PER_THREAD (EPS_E_FLOATS / (BLOCK * 4))  // 5

// Fast numerically-stable softplus using hardware transcendentals:
//   softplus(x) = max(x,0) + log(1 + exp(-|x|))
__device__ __forceinline__ float fast_softplus(float x) {
    const float e = __expf(-fabsf(x));          // v_exp_f32 path
    return fmaxf(x, 0.0f) + __logf(1.0f + e);   // v_log_f32 path
}

__device__ __forceinline__ unsigned lds_addr_of(const void* p) {
    // Generic (flat) LDS pointer: low 32 bits are the LDS byte offset.
    return (unsigned)(size_t)p;
}

// CDNA5 async global->LDS copy, GVS (saddr) form. Per the ISA, INST_OFFSET is
// added to BOTH the LDS address (VDST) and the global address (SADDR+VADDR),
// so one 32-bit lane offset serves every granule.
template <int IMM_OFF>
__device__ __forceinline__ void async_b128_to_lds(unsigned lds_byte_addr,
                                                  unsigned lane_byte_off,
                                                  const void* sgpr_base) {
    asm volatile("global_load_async_to_lds_b128 %0, %1, %2 offset:%3"
                 :: "v"(lds_byte_addr), "v"(lane_byte_off), "s"(sgpr_base), "n"(IMM_OFF)
                 : "memory");
}

__global__ __launch_bounds__(BLOCK) void vfm_fused_kernel(
    const int*   __restrict__ x,            // (B,2) int32 indices, flat rows
    const float* __restrict__ bias_table,   // (VOCAB,2)
    const float* __restrict__ entity_table, // (VOCAB,2*D)
    const float* __restrict__ alpha,        // (1,)
    const float* __restrict__ global_bias,  // (1,)
    const float* __restrict__ eps_bias,     // (1,2B)
    const float* __restrict__ eps_entity,   // (1,2B,D)
    float*       __restrict__ out,          // pred | std_dev | kl_bias | kl_entity
    int n_pairs)
{
    __shared__ float s_eps_e[EPS_E_FLOATS]; // 20 KB
    __shared__ float s_eps_b[BLOCK];        //  1 KB

    const int t    = threadIdx.x;
    const int row0 = blockIdx.x * BLOCK;
    const int r    = row0 + t;              // this thread's flat row

    // ---------- Phase 1: async eps staging (coalesced 16B/lane, ASYNCcnt) ----------
    {
        const float*   gbase = eps_entity + (size_t)row0 * DD;
        const unsigned voff  = (unsigned)t * 16u;
        const unsigned lbase = lds_addr_of(s_eps_e) + voff;
        async_b128_to_lds<0 * 4096>(lbase, voff, gbase);
        async_b128_to_lds<1 * 4096>(lbase, voff, gbase);
        async_b128_to_lds<2 * 4096>(lbase, voff, gbase);
        async_b128_to_lds<3 * 4096>(lbase, voff, gbase);
        async_b128_to_lds<4 * 4096>(lbase, voff, gbase);
        if (t < BLOCK / 4) {                // 256 floats = 64 x b128
            async_b128_to_lds<0>(lds_addr_of(s_eps_b) + voff, voff, eps_bias + row0);
        }
    }

    // ---------- Phase 2: random gathers (overlap with async DMA) ----------
    const int idx = x[r];

    const float2 bb = *(const float2*)(bias_table + (size_t)idx * 2);

    alignas(16) float f[2 * DD];            // [mu(20) | raw_s(20)] for this row
    {
        const float4* ep = (const float4*)(entity_table + (size_t)idx * (2 * DD));
        #pragma unroll
        for (int i = 0; i < 10; ++i) {      // 160B contiguous row
            const float4 v = ep[i];
            f[4*i+0] = v.x; f[4*i+1] = v.y; f[4*i+2] = v.z; f[4*i+3] = v.w;
        }
    }

    // eps-independent bias math while DMA + gathers are in flight
    const float mu_b = bb.x;
    const float sb   = fast_softplus(bb.y);
    const float klb  = -__logf(sb) + 0.5f * fmaf(sb, sb, mu_b * mu_b) - 0.5f;

    // ---------- Phase 3: wait on this wave's async copies, then barrier ----------
    asm volatile("s_wait_asynccnt 0x0" ::: "memory");
    __syncthreads();

    const float* eps  = &s_eps_e[t * DD];
    const float  epsb = s_eps_b[t];
    const float  bias = fmaf(sb, epsb, mu_b);

    // ---------- Phase 4: fused reparam + KL + cross-lane pair dot ----------
    // Rows 2b and 2b+1 live in adjacent lanes of the same wave32; both lanes
    // accumulate the identical pair dot via lane-xor-1 exchange.
    float kle = 0.0f, dot = 0.0f;
    #pragma unroll
    for (int d = 0; d < DD; ++d) {
        const float mu = f[d];
        const float s  = fast_softplus(f[DD + d]);
        const float e  = fmaf(s, eps[d], mu);
        kle += -__logf(s) + 0.5f * fmaf(s, s, mu * mu) - 0.5f;
        const float e_partner = __shfl_xor(e, 1, 32);
        dot = fmaf(e, e_partner, dot);
    }

    // ---------- Phase 5: outputs ----------
    float* pred    = out;                   // [0, n_pairs)
    float* stddev  = out + n_pairs;         // [n_pairs]
    float* kl_bias = out + n_pairs + 1;     // 2*n_pairs elems
    float* kl_ent  = kl_bias + 2 * n_pairs; // 2*n_pairs elems

    kl_bias[r] = klb;                       // fully coalesced b32 stores
    kl_ent[r]  = kle;

    const float bias_partner = __shfl_xor(bias, 1, 32);
    if ((t & 1) == 0) {                     // even lane owns the pair
        pred[r >> 1] = global_bias[0] + bias + bias_partner + dot;
    }
    if (r == 0) {
        stddev[0] = sqrtf(1.0f / fast_softplus(alpha[0]));
    }
}

extern "C" void kernel_launch(void* const* d_in, const int* in_sizes, int n_in,
                              void* d_out, int out_size, void* d_ws, size_t ws_size,
                              hipStream_t stream) {
    (void)n_in; (void)out_size; (void)d_ws; (void)ws_size;
    const int*   x            = (const int*)d_in[0];
    const float* bias_table   = (const float*)d_in[1];
    const float* entity_table = (const float*)d_in[2];
    const float* alpha        = (const float*)d_in[3];
    const float* global_bias  = (const float*)d_in[4];
    const float* eps_bias     = (const float*)d_in[5];
    const float* eps_entity   = (const float*)d_in[6];
    float* out = (float*)d_out;

    const int n_rows  = in_sizes[0];        // 2B = 131072
    const int n_pairs = n_rows / 2;         // 65536
    const int grid    = n_rows / BLOCK;     // 512 blocks of 256 threads

    vfm_fused_kernel<<<grid, BLOCK, 0, stream>>>(
        x, bias_table, entity_table, alpha, global_bias, eps_bias, eps_entity,
        out, n_pairs);
}